// MCMHEdgeEncoder_69681549410497
// MI455X (gfx1250) — compile-verified
//
#include <hip/hip_runtime.h>

// ---------------------------------------------------------------------------
// MCMHEdgeEncoder for MI455X (gfx1250, wave32)
//
//   P[n, 0: 8] = X[n,:] . W1[c,:]
//   P[n, 8:16] = X[n,:] . W2[c-8,:]
//   out[e, c]  = P[e0[e], c] + P[e1[e], 8+c]      c in 0..7
//
// Roofline: 0.82 GFLOP (negligible) vs ~360 MB DRAM traffic
//   (X 102.4 MB + edge_index 51.2 MB + out 204.8 MB) -> ~15.5 us @ 23.3 TB/s.
// Kernel 1: fp32 WMMA projection [N,256]x[256,16] (exact vs the fp32 ref).
// Kernel 2: gather-add; P (6.4 MB) is L2-resident (192 MB L2), streams use
//           non-temporal hints so they don't evict P.
// ---------------------------------------------------------------------------

typedef __attribute__((ext_vector_type(2))) float v2f;
typedef __attribute__((ext_vector_type(4))) float v4f;
typedef __attribute__((ext_vector_type(8))) float v8f;

#define IN_CH  256
#define OUT_CH 8
#define P_COLS 16   // [ P1 | P2 ] packed per node row

// --------------------------------------------------------------------------
// Kernel 1: P = X @ [W1;W2]^T using V_WMMA_F32_16X16X4_F32.
// One wave per 16-row tile of X, 64 WMMA steps over K=256.
//
// ISA VGPR layouts (CDNA5 ISA):
//   A (16x4 f32, 2 VGPRs): lane half h, vgpr j -> K = 2h + j, M = lane&15
//   B (4x16 f32, 2 VGPRs): lane half h, vgpr j -> K = 2h + j, N = lane&15
//   C/D (16x16 f32, 8 VGPRs): vgpr j -> M = j + 8h, N = lane&15
// --------------------------------------------------------------------------
__global__ void proj_wmma_f32(const float* __restrict__ X,
                              const float* __restrict__ W1,
                              const float* __restrict__ W2,
                              float* __restrict__ P,
                              int nrows) {
  const int wave = (int)((blockIdx.x * blockDim.x + threadIdx.x) >> 5);
  const int lane = (int)(threadIdx.x & 31);
  const int r0   = wave * 16;
  if (r0 >= nrows) return;            // wave-uniform: EXEC stays all-ones

  const int half = lane >> 4;         // 0 -> K{0,1}, 1 -> K{2,3}
  const int l15  = lane & 15;

  // Clamp row for a possible ragged last tile (keeps EXEC all-ones for WMMA).
  const int arow = (r0 + l15 < nrows) ? (r0 + l15) : (nrows - 1);
  const float* xfrag = X + (size_t)arow * IN_CH + 2 * half;

  // Virtual B row n: W1 row n for n<8, W2 row n-8 for n>=8.
  const float* wfrag =
      (l15 < OUT_CH ? (W1 + (size_t)l15 * IN_CH)
                    : (W2 + (size_t)(l15 - OUT_CH) * IN_CH)) + 2 * half;

  v8f acc = {};
#pragma unroll 8
  for (int k4 = 0; k4 < IN_CH; k4 += 4) {
    v2f a = *(const v2f*)(xfrag + k4);   // global_load_b64 (coalesces per row)
    v2f b = *(const v2f*)(wfrag + k4);   // 16 KB W total: WGP$/L2 hits
    // (neg_a, A, neg_b, B, c_mod, C, reuse_a, reuse_b)
    acc = __builtin_amdgcn_wmma_f32_16x16x4_f32(false, a, false, b,
                                                (short)0, acc, false, false);
  }

  // Scatter D: row = r0 + 8*half + j, col = l15.
  const int rbase = r0 + half * 8;
  float* prow = P + (size_t)rbase * P_COLS + l15;
  if (r0 + 16 <= nrows) {
    // Full tile (always taken for N % 16 == 0): wave-uniform, no exec churn.
#pragma unroll
    for (int j = 0; j < 8; ++j)
      prow[(size_t)j * P_COLS] = acc[j];
  } else {
    // Ragged last tile: per-row guard.
#pragma unroll
    for (int j = 0; j < 8; ++j)
      if (rbase + j < nrows) prow[(size_t)j * P_COLS] = acc[j];
  }
}

// --------------------------------------------------------------------------
// Kernel 2: out[e,:] = P[src, 0:8] + P[dst, 8:16]
// One thread per edge. 2x32B gathers (L2-resident P), 32B coalesced NT store.
// --------------------------------------------------------------------------
__global__ void gather_add_edges(const int* __restrict__ ei,   // [2, E]
                                 const float* __restrict__ P,
                                 float* __restrict__ out,
                                 int nedges) {
  const int e = (int)(blockIdx.x * blockDim.x + threadIdx.x);
  if (e >= nedges) return;

  const int src = __builtin_nontemporal_load(ei + e);                  // e0
  const int dst = __builtin_nontemporal_load(ei + (size_t)nedges + e); // e1

  const v4f* p1 = (const v4f*)(P + (size_t)src * P_COLS);        // cols 0..7
  const v4f* p2 = (const v4f*)(P + (size_t)dst * P_COLS + 8);    // cols 8..15

  v4f r0 = p1[0] + p2[0];
  v4f r1 = p1[1] + p2[1];

  v4f* o = (v4f*)(out + (size_t)e * OUT_CH);
  __builtin_nontemporal_store(r0, o);      // 205 MB stream: keep out of L2
  __builtin_nontemporal_store(r1, o + 1);
}

// --------------------------------------------------------------------------
// Host-side launch (graph-capture safe: only kernel launches on `stream`).
//   d_in[0] = X           float32 [N, 256]
//   d_in[1] = edge_index  int32   [2, E]
//   d_in[2] = W1          float32 [8, 256]
//   d_in[3] = W2          float32 [8, 256]
//   d_out   = float32 [E, 8]
//   d_ws    = P scratch, needs N*16*4 = 6.4 MB
// --------------------------------------------------------------------------
extern "C" void kernel_launch(void* const* d_in, const int* in_sizes, int n_in,
                              void* d_out, int out_size, void* d_ws, size_t ws_size,
                              hipStream_t stream) {
  const float* X  = (const float*)d_in[0];
  const int*   ei = (const int*)d_in[1];
  const float* W1 = (const float*)d_in[2];
  const float* W2 = (const float*)d_in[3];

  const int nrows  = in_sizes[0] / IN_CH;   // 100000
  const int nedges = in_sizes[1] / 2;       // 6400000

  float* P   = (float*)d_ws;                // [nrows, 16]
  float* out = (float*)d_out;               // [nedges, 8]

  // Kernel 1: one wave per 16-row tile; 256 threads = 8 waves per block.
  const int ntiles  = (nrows + 15) / 16;    // 6250
  const int blocks1 = (ntiles + 7) / 8;     // 782
  proj_wmma_f32<<<blocks1, 256, 0, stream>>>(X, W1, W2, P, nrows);

  // Kernel 2: one thread per edge.
  const int blocks2 = (nedges + 255) / 256; // 25000
  gather_add_edges<<<blocks2, 256, 0, stream>>>(ei, P, out, nedges);
}